// GaussPerturber_68977174774014
// MI455X (gfx1250) — compile-verified
//
#include <hip/hip_runtime.h>
#include <math.h>

typedef __attribute__((ext_vector_type(2))) float v2f;
typedef __attribute__((ext_vector_type(8))) float v8f;

#define IN_DIM 768
#define HID    1024
#define OUT_DIM 18
#define NB     16
#define PERTSZ 64
#define NCH    3
#define NFR    2
#define IMH    512
#define IMW    512

__device__ __forceinline__ float wave_reduce_add(float v) {
#pragma unroll
    for (int off = 16; off > 0; off >>= 1) v += __shfl_xor(v, off, 32);
    return v;
}

// ---------------------------------------------------------------------------
// Kernel 1: whole MLP in one workgroup (32 waves), f32 WMMA 16x16x4.
// M=16 rows == one WMMA tile; 64 N-tiles per hidden layer -> 2 tiles/wave.
// ---------------------------------------------------------------------------
__global__ __launch_bounds__(1024) void mlp_wmma_kernel(
    const float* __restrict__ enc,
    const float* __restrict__ W1, const float* __restrict__ b1,
    const float* __restrict__ g1, const float* __restrict__ be1,
    const float* __restrict__ W2, const float* __restrict__ b2,
    const float* __restrict__ g2, const float* __restrict__ be2,
    const float* __restrict__ W3, const float* __restrict__ b3,
    const float* __restrict__ g3, const float* __restrict__ be3,
    float* __restrict__ params)
{
    __shared__ float xbuf[NB * HID];          // 64 KB activation buffer
    const int tid  = threadIdx.x;
    const int wave = tid >> 5;
    const int lane = tid & 31;
    const int m    = lane & 15;               // A row / B,C column within tile
    const int half = lane >> 4;               // 0: K={0,1},M_base=0 ; 1: K={2,3},M_base=8
    const int kofs = half * 2;
    const int rowO = half * 8;

    // ----- Layer 1: enc(16x768) @ W1(768x1024) -----
    {
        const int n0 = wave * 16 + m;
        const int n1 = (wave + 32) * 16 + m;
        v8f acc0 = {}; v8f acc1 = {};
        for (int k0 = 0; k0 < IN_DIM; k0 += 4) {
            const int ka = k0 + kofs;
            v2f a, bA, bB;
            a.x  = enc[m * IN_DIM + ka];
            a.y  = enc[m * IN_DIM + ka + 1];
            bA.x = W1[ka * HID + n0];       bA.y = W1[(ka + 1) * HID + n0];
            bB.x = W1[ka * HID + n1];       bB.y = W1[(ka + 1) * HID + n1];
            acc0 = __builtin_amdgcn_wmma_f32_16x16x4_f32(false, a, false, bA, (short)0, acc0, false, false);
            acc1 = __builtin_amdgcn_wmma_f32_16x16x4_f32(false, a, false, bB, (short)0, acc1, false, false);
        }
#pragma unroll
        for (int r = 0; r < 8; ++r) {
            xbuf[(rowO + r) * HID + n0] = acc0[r] + b1[n0];
            xbuf[(rowO + r) * HID + n1] = acc1[r] + b1[n1];
        }
    }
    __syncthreads();

    // ----- LayerNorm + ReLU (waves 0..15, one row each, shuffle reduce) -----
    if (wave < NB) {
        float* row = &xbuf[wave * HID];
        float s = 0.f, s2 = 0.f;
        for (int j = lane; j < HID; j += 32) { float v = row[j]; s += v; s2 += v * v; }
        s  = wave_reduce_add(s);
        s2 = wave_reduce_add(s2);
        const float mean = s * (1.0f / HID);
        const float var  = s2 * (1.0f / HID) - mean * mean;
        const float inv  = rsqrtf(var + 1e-5f);
        for (int j = lane; j < HID; j += 32) {
            float v = (row[j] - mean) * inv * g1[j] + be1[j];
            row[j] = v > 0.f ? v : 0.f;
        }
    }
    __syncthreads();

    // ----- Layer 2: x(16x1024) @ W2(1024x1024), A from LDS -----
    {
        const int n0 = wave * 16 + m;
        const int n1 = (wave + 32) * 16 + m;
        v8f acc0 = {}; v8f acc1 = {};
        for (int k0 = 0; k0 < HID; k0 += 4) {
            const int ka = k0 + kofs;
            v2f a, bA, bB;
            a.x  = xbuf[m * HID + ka];
            a.y  = xbuf[m * HID + ka + 1];
            bA.x = W2[ka * HID + n0];       bA.y = W2[(ka + 1) * HID + n0];
            bB.x = W2[ka * HID + n1];       bB.y = W2[(ka + 1) * HID + n1];
            acc0 = __builtin_amdgcn_wmma_f32_16x16x4_f32(false, a, false, bA, (short)0, acc0, false, false);
            acc1 = __builtin_amdgcn_wmma_f32_16x16x4_f32(false, a, false, bB, (short)0, acc1, false, false);
        }
        __syncthreads();   // all reads of xbuf done before overwrite
#pragma unroll
        for (int r = 0; r < 8; ++r) {
            xbuf[(rowO + r) * HID + n0] = acc0[r] + b2[n0];
            xbuf[(rowO + r) * HID + n1] = acc1[r] + b2[n1];
        }
    }
    __syncthreads();

    if (wave < NB) {
        float* row = &xbuf[wave * HID];
        float s = 0.f, s2 = 0.f;
        for (int j = lane; j < HID; j += 32) { float v = row[j]; s += v; s2 += v * v; }
        s  = wave_reduce_add(s);
        s2 = wave_reduce_add(s2);
        const float mean = s * (1.0f / HID);
        const float var  = s2 * (1.0f / HID) - mean * mean;
        const float inv  = rsqrtf(var + 1e-5f);
        for (int j = lane; j < HID; j += 32) {
            float v = (row[j] - mean) * inv * g2[j] + be2[j];
            row[j] = v > 0.f ? v : 0.f;
        }
    }
    __syncthreads();

    // ----- Layer 3: x(16x1024) @ W3(1024x18); waves 0,1 only (N padded to 32) -----
    {
        v8f acc = {};
        const int nb = wave * 16 + m;
        const bool active = (wave < 2);
        if (active) {
            const bool valid = nb < OUT_DIM;
            for (int k0 = 0; k0 < HID; k0 += 4) {
                const int ka = k0 + kofs;
                v2f a, bfr;
                a.x = xbuf[m * HID + ka];
                a.y = xbuf[m * HID + ka + 1];
                bfr.x = valid ? W3[ka * OUT_DIM + nb] : 0.f;
                bfr.y = valid ? W3[(ka + 1) * OUT_DIM + nb] : 0.f;
                acc = __builtin_amdgcn_wmma_f32_16x16x4_f32(false, a, false, bfr, (short)0, acc, false, false);
            }
        }
        __syncthreads();   // layer-3 reads of xbuf complete
        if (active) {
#pragma unroll
            for (int r = 0; r < 8; ++r)
                xbuf[(rowO + r) * 32 + nb] = acc[r] + (nb < OUT_DIM ? b3[nb] : 0.f);
        }
    }
    __syncthreads();

    // ----- LN(18) + tanh, one row per thread -----
    if (tid < NB) {
        const float* row = &xbuf[tid * 32];
        float s = 0.f, s2 = 0.f;
        for (int j = 0; j < OUT_DIM; ++j) { float v = row[j]; s += v; s2 += v * v; }
        const float mean = s * (1.0f / OUT_DIM);
        const float var  = s2 * (1.0f / OUT_DIM) - mean * mean;
        const float inv  = rsqrtf(var + 1e-5f);
        for (int j = 0; j < OUT_DIM; ++j) {
            float v = (row[j] - mean) * inv * g3[j] + be3[j];
            params[tid * OUT_DIM + j] = tanhf(v);
        }
    }
}

// ---------------------------------------------------------------------------
// Kernel 2: Gaussian patches. scale=amp/max(dens) cancels all constants:
// pert = amp * exp(-0.5*(quad - min(quad))).
// ---------------------------------------------------------------------------
__global__ __launch_bounds__(256) void gauss_kernel(const float* __restrict__ params,
                                                    float* __restrict__ pert)
{
    __shared__ float red[256];
    const int bc = blockIdx.x;                // 0..47
    const int b = bc / NCH, c = bc % NCH;
    const float* p = &params[b * OUT_DIM + c * 6];
    const float mean0 = p[0] * (PERTSZ * 0.5f);
    const float mean1 = p[1] * (PERTSZ * 0.5f);
    const float rs = (p[2] + 1.f) * ((PERTSZ * 0.5f - 1.f) * 0.5f) + 1.f;
    const float cs = (p[3] + 1.f) * ((PERTSZ * 0.5f - 1.f) * 0.5f) + 1.f;
    const float corr = p[4] * 0.98f;
    const float amp  = p[5];
    const float det = (rs * cs) * (rs * cs) * (1.f - corr * corr);
    const float i00 = cs * cs / det;
    const float i11 = rs * rs / det;
    const float i01 = -(rs * cs * corr) / det;

    const int tid = threadIdx.x;
    const float step = (float)PERTSZ / (float)(PERTSZ - 1);
    float quad[16];
    float qmin = 3.4e38f;
#pragma unroll
    for (int i = 0; i < 16; ++i) {
        const int pt = tid + i * 256;
        const int gi = pt >> 6, gj = pt & 63;
        const float d0 = (-(PERTSZ * 0.5f) + gi * step) - mean0;
        const float d1 = (-(PERTSZ * 0.5f) + gj * step) - mean1;
        const float q = i00 * d0 * d0 + 2.f * i01 * d0 * d1 + i11 * d1 * d1;
        quad[i] = q;
        qmin = fminf(qmin, q);
    }
    red[tid] = qmin;
    __syncthreads();
    for (int s = 128; s > 0; s >>= 1) {
        if (tid < s) red[tid] = fminf(red[tid], red[tid + s]);
        __syncthreads();
    }
    const float q0 = red[0];
    float* out = &pert[(size_t)bc * (PERTSZ * PERTSZ)];
#pragma unroll
    for (int i = 0; i < 16; ++i) {
        const int pt = tid + i * 256;
        out[pt] = amp * expf(-0.5f * (quad[i] - q0));
    }
}

// ---------------------------------------------------------------------------
// Kernel 3: bulk video copy (dominant cost), float4 coalesced.
// ---------------------------------------------------------------------------
__global__ void copy_kernel(const float4* __restrict__ in, float4* __restrict__ out, int n4)
{
    int idx    = blockIdx.x * blockDim.x + threadIdx.x;
    int stride = gridDim.x * blockDim.x;
    for (int i = idx; i < n4; i += stride) out[i] = in[i];
}

// ---------------------------------------------------------------------------
// Kernel 4: scatter-add 64x64 patches into frame 0. Regions are disjoint
// per (b,c) -> no atomics needed.
// ---------------------------------------------------------------------------
__global__ __launch_bounds__(256) void patch_add_kernel(const float* __restrict__ pert,
                                                        const int* __restrict__ loc,
                                                        float* __restrict__ out)
{
    const int bc = blockIdx.x;
    const int b = bc / NCH, c = bc % NCH;
    const int h0 = loc[b * 2 + 0] - PERTSZ / 2;
    const int w0 = loc[b * 2 + 1] - PERTSZ / 2;
    float* frame = &out[(size_t)((b * NCH + c) * NFR + 0) * IMH * IMW];
    const float* pp = &pert[(size_t)bc * (PERTSZ * PERTSZ)];
    for (int i = threadIdx.x; i < PERTSZ * PERTSZ; i += 256) {
        const int pi = i >> 6, pj = i & 63;
        const int h = h0 + pi, w = w0 + pj;
        if (h >= 0 && h < IMH && w >= 0 && w < IMW)
            frame[h * IMW + w] += pp[i];
    }
}

extern "C" void kernel_launch(void* const* d_in, const int* in_sizes, int n_in,
                              void* d_out, int out_size, void* d_ws, size_t ws_size,
                              hipStream_t stream) {
    (void)in_sizes; (void)n_in; (void)ws_size;
    // setup_inputs() dict order:
    const float* video = (const float*)d_in[0];
    const float* enc   = (const float*)d_in[1];
    const int*   loc   = (const int*)  d_in[2];
    const float* W1 = (const float*)d_in[3];
    const float* b1 = (const float*)d_in[4];
    const float* g1 = (const float*)d_in[5];
    const float* be1= (const float*)d_in[6];
    const float* W2 = (const float*)d_in[7];
    const float* b2 = (const float*)d_in[8];
    const float* g2 = (const float*)d_in[9];
    const float* be2= (const float*)d_in[10];
    const float* W3 = (const float*)d_in[11];
    const float* b3 = (const float*)d_in[12];
    const float* g3 = (const float*)d_in[13];
    const float* be3= (const float*)d_in[14];
    float* out = (float*)d_out;

    float* ws_f   = (float*)d_ws;
    float* params = ws_f;            // 16*18 floats
    float* pert   = ws_f + 1024;     // 48*64*64 floats

    // 1) MLP -> pert params (one workgroup, WMMA f32)
    mlp_wmma_kernel<<<1, 1024, 0, stream>>>(enc, W1, b1, g1, be1,
                                            W2, b2, g2, be2,
                                            W3, b3, g3, be3, params);
    // 2) Gaussian patches
    gauss_kernel<<<48, 256, 0, stream>>>(params, pert);
    // 3) Bulk video copy (dominant: ~200 MB traffic)
    const int n4 = out_size / 4;     // 25,165,824 / 4 float4s
    copy_kernel<<<4096, 256, 0, stream>>>((const float4*)video, (float4*)out, n4);
    // 4) Add patches to frame 0
    patch_add_kernel<<<48, 256, 0, stream>>>(pert, loc, out);
}